// sysAdminModel_80066780332528
// MI455X (gfx1250) — compile-verified
//
#include <hip/hip_runtime.h>

typedef __attribute__((ext_vector_type(2))) float v2f;
typedef __attribute__((ext_vector_type(8))) float v8f;

#define GEMM_WAVES 4

// ---------------------------------------------------------------------------
// K1: zero the workspace (agg + cnt). ws is poisoned by the harness, so this
// must run every launch.
// ---------------------------------------------------------------------------
__global__ void zero_ws_kernel(float* __restrict__ ws, long long n) {
  long long i = (long long)blockIdx.x * blockDim.x + threadIdx.x;
  long long stride = (long long)gridDim.x * blockDim.x;
  for (; i < n; i += stride) ws[i] = 0.0f;
}

// ---------------------------------------------------------------------------
// K2: wave-per-edge scatter add.  32 lanes x float4 = 128 channels.
// x and agg (25.6 MB each) are L2-resident on MI455X (192 MB L2), so the
// random-index atomics resolve in L2.  atomicAdd result unused -> emits
// non-returning global_atomic_add_f32 (STOREcnt path).
// ---------------------------------------------------------------------------
__global__ void edge_scatter_kernel(const float* __restrict__ x,
                                    const int* __restrict__ ei,
                                    float* __restrict__ agg,
                                    float* __restrict__ cnt,
                                    int E) {
  int wave = blockIdx.x * (blockDim.x >> 5) + (threadIdx.x >> 5);
  if (wave >= E) return;
  int lane = threadIdx.x & 31;
  int src = ei[wave];
  int dst = ei[E + wave];
  const float4* xv = (const float4*)(x + (size_t)src * 128);
  float4 v = xv[lane];
  float* ap = agg + (size_t)dst * 128 + lane * 4;
  atomicAdd(ap + 0, v.x);
  atomicAdd(ap + 1, v.y);
  atomicAdd(ap + 2, v.z);
  atomicAdd(ap + 3, v.w);
  if (lane == 0) atomicAdd(cnt + dst, 1.0f);
}

// ---------------------------------------------------------------------------
// K3: fused SAGE layer.  One wave owns a 16-row tile.
//   h   = relu(mean @ W_l + b_l + x @ W_r)   (both GEMMs share accumulators)
//   out = h @ W2 + b2
// Matrix math via V_WMMA_F32_16X16X4_F32 (fp32 in/accum, matches reference).
// W2 (32 KB, shared by every wave) is staged into LDS once per workgroup via
// GLOBAL_LOAD_ASYNC_TO_LDS_B128 (ASYNCcnt path), before any divergence.
// ---------------------------------------------------------------------------
__global__ __launch_bounds__(GEMM_WAVES * 32)
void sage_fused_kernel(const float* __restrict__ x,
                       const float* __restrict__ agg,
                       const float* __restrict__ cnt,
                       const float* __restrict__ Wl,
                       const float* __restrict__ bl,
                       const float* __restrict__ Wr,
                       const float* __restrict__ W2,
                       const float* __restrict__ b2,
                       float* __restrict__ out,
                       int ntiles) {
  __shared__ float hbuf[GEMM_WAVES][16 * 128];
  __shared__ float w2s[128 * 64];   // W2 staged row-major, 32 KB

  // ---- async-stage W2 into LDS (all 128 threads, EXEC all-1s) ----
  {
    int t = threadIdx.x;                       // 0..127
    int base = t * 64;                         // 64 floats per thread
    unsigned lds0 = (unsigned)(size_t)(&w2s[0]);
#pragma unroll
    for (int j = 0; j < 16; ++j) {             // 16 x B128 = 64 floats
      const float* gp = W2 + base + j * 4;
      unsigned la = lds0 + (unsigned)(base + j * 4) * 4u;
      asm volatile("global_load_async_to_lds_b128 %0, %1, off"
                   :: "v"(la), "v"(gp) : "memory");
    }
    asm volatile("s_wait_asynccnt 0x0" ::: "memory");
  }
  __syncthreads();                             // publish w2s to all waves

  int lane   = threadIdx.x & 31;
  int wid    = threadIdx.x >> 5;
  int laneLo = lane & 15;
  int hi2    = (lane >> 4) << 1;  // 0 for lanes 0-15, 2 for lanes 16-31
  int rbase  = (lane >> 4) << 3;  // 0 or 8: C/D row offset for upper half-wave

  int tile = blockIdx.x * GEMM_WAVES + wid;
  if (tile >= ntiles) return;     // wave-granular: EXEC stays all-1s for WMMA
  int row0 = tile * 16;
  int arow = row0 + laneLo;       // A-matrix row owned by this lane

  float invc = 1.0f / fmaxf(cnt[arow], 1.0f);

  v8f acc[8];
#pragma unroll
  for (int i = 0; i < 8; ++i) acc[i] = (v8f)0.0f;

  const float* xrow = x   + (size_t)arow * 128;
  const float* grow = agg + (size_t)arow * 128;

  // GEMM 1+2 fused: acc = mean@Wl + x@Wr  (16x128 tile, 8 N-subtiles)
  for (int kt = 0; kt < 32; ++kt) {
    int kk = kt * 4 + hi2;
    v2f ax = *(const v2f*)(xrow + kk);
    v2f ag = *(const v2f*)(grow + kk);
    v2f am;
    am.x = ag.x * invc;
    am.y = ag.y * invc;
#pragma unroll
    for (int nt = 0; nt < 8; ++nt) {
      int n = nt * 16 + laneLo;
      v2f bL, bR;
      bL.x = Wl[(kk + 0) * 128 + n];
      bL.y = Wl[(kk + 1) * 128 + n];
      bR.x = Wr[(kk + 0) * 128 + n];
      bR.y = Wr[(kk + 1) * 128 + n];
      acc[nt] = __builtin_amdgcn_wmma_f32_16x16x4_f32(
          false, am, false, bL, (short)0, acc[nt], false, false);
      acc[nt] = __builtin_amdgcn_wmma_f32_16x16x4_f32(
          false, ax, false, bR, (short)0, acc[nt], false, false);
    }
  }

  // bias + relu, spill h tile to LDS in row-major so we can re-fragment it
  // as an A-matrix (C-layout has N across lanes; A needs M across lanes).
#pragma unroll
  for (int nt = 0; nt < 8; ++nt) {
    float bias = bl[nt * 16 + laneLo];
#pragma unroll
    for (int v = 0; v < 8; ++v) {
      float hv = fmaxf(acc[nt][v] + bias, 0.0f);
      hbuf[wid][(rbase + v) * 128 + nt * 16 + laneLo] = hv;
    }
  }
  // Per-wave LDS RAW: DS ops are in-order within a wave; this barrier only
  // pins the compiler's schedule (stores before dependent cross-lane loads).
  __builtin_amdgcn_wave_barrier();

  // GEMM 3: out = h @ W2 + b2   (16x64 tile, 4 N-subtiles), B from LDS
  v8f acc2[4];
#pragma unroll
  for (int i = 0; i < 4; ++i) acc2[i] = (v8f)0.0f;

  for (int kt = 0; kt < 32; ++kt) {
    int kk = kt * 4 + hi2;
    v2f a;
    a.x = hbuf[wid][laneLo * 128 + kk + 0];
    a.y = hbuf[wid][laneLo * 128 + kk + 1];
#pragma unroll
    for (int nt = 0; nt < 4; ++nt) {
      int n = nt * 16 + laneLo;
      v2f b;
      b.x = w2s[(kk + 0) * 64 + n];
      b.y = w2s[(kk + 1) * 64 + n];
      acc2[nt] = __builtin_amdgcn_wmma_f32_16x16x4_f32(
          false, a, false, b, (short)0, acc2[nt], false, false);
    }
  }

#pragma unroll
  for (int nt = 0; nt < 4; ++nt) {
    float bias = b2[nt * 16 + laneLo];
#pragma unroll
    for (int v = 0; v < 8; ++v) {
      int r = row0 + rbase + v;
      out[(size_t)r * 64 + nt * 16 + laneLo] = acc2[nt][v] + bias;
    }
  }
}

// ---------------------------------------------------------------------------
extern "C" void kernel_launch(void* const* d_in, const int* in_sizes, int n_in,
                              void* d_out, int out_size, void* d_ws, size_t ws_size,
                              hipStream_t stream) {
  const float* x  = (const float*)d_in[0];
  const int*   ei = (const int*)  d_in[1];   // edge_index [2, E] (int32)
  const float* Wl = (const float*)d_in[2];
  const float* bl = (const float*)d_in[3];
  const float* Wr = (const float*)d_in[4];
  const float* W2 = (const float*)d_in[5];
  const float* b2 = (const float*)d_in[6];
  float* out = (float*)d_out;

  int n = in_sizes[0] / 128;   // 50000 nodes
  int E = in_sizes[1] / 2;     // 1.6M edges

  float* agg = (float*)d_ws;                 // [n, 128]
  float* cnt = agg + (size_t)n * 128;        // [n]

  long long zn = (long long)n * 128 + n;
  zero_ws_kernel<<<1024, 256, 0, stream>>>(agg, zn);

  const int wavesPerBlk = 8;                 // 256 threads
  int blocksB = (E + wavesPerBlk - 1) / wavesPerBlk;
  edge_scatter_kernel<<<blocksB, wavesPerBlk * 32, 0, stream>>>(x, ei, agg, cnt, E);

  int ntiles  = n / 16;                      // 50000 % 16 == 0 -> exact
  int blocksC = (ntiles + GEMM_WAVES - 1) / GEMM_WAVES;
  sage_fused_kernel<<<blocksC, GEMM_WAVES * 32, 0, stream>>>(
      x, agg, cnt, Wl, bl, Wr, W2, b2, out, ntiles);
}